// SparseInteractionAdditiveRiskModel_44753559225099
// MI455X (gfx1250) — compile-verified
//
#include <hip/hip_runtime.h>
#include <math.h>

// ---------------------------------------------------------------------------
// SparseInteractionAdditiveRiskModel for MI455X (gfx1250, wave32)
//
// Roofline: 23 MB of HBM traffic (~1us) vs B*1024 exact erf-GELU evals
// (~1e10 VALU lane-ops) => erf-bound, fp32 mandatory for accuracy.
//
// WMMA use: V_WMMA_F32_16X16X4_F32 (bit-exact fp32 FMA, RNE) performs the
// 1024-long per-row weighted reductions. One wave owns 16 rows; each WMMA
// consumes a K-slice of 4 weighted-gelu values (2 per lane, exactly the
// documented 16x4 f32 A layout) against B == all-ones, accumulating row sums
// in the 16x16 f32 C tile. B == 1 makes the result independent of the
// (undocumented) f32 B-matrix K striping: only the documented A and C
// layouts matter. No reduction FMAs, no cross-lane shuffles, and every gelu
// is computed exactly once across the wave.
// ---------------------------------------------------------------------------

typedef __attribute__((ext_vector_type(2))) float v2f;
typedef __attribute__((ext_vector_type(4))) float v4f;
typedef __attribute__((ext_vector_type(8))) float v8f;

#define NF 10
#define NH 64
#define NP 6
#define KADD (NF * NH)   /* 640 */
#define KPAIR (NP * NH)  /* 384 */

// weighted exact GELU: gelu(a) * w2  with  hw = 0.5*w2 pre-folded.
//   gelu(a)*w2 = 0.5*a*w2*(1+erf(a/sqrt2)) = t + t*erf(z),  t = a*hw
__device__ __forceinline__ float wgelu(float a, float hw) {
    float t = a * hw;
    float e = erff(a * 0.70710678118654752440f);
    return fmaf(t, e, t);
}

__global__ __launch_bounds__(256) void risk_model_kernel(
    const float* __restrict__ x,
    const float* __restrict__ raw_mean, const float* __restrict__ raw_std,
    const float* __restrict__ fw1, const float* __restrict__ fb1,
    const float* __restrict__ fw2, const float* __restrict__ fb2,
    const float* __restrict__ pw1, const float* __restrict__ pb1,
    const float* __restrict__ pw2, const float* __restrict__ pb2,
    float* __restrict__ out, int nrows)
{
    // ---- stage packed weights into LDS (16.4 KB of 320 KB/WGP) ----
    __shared__ v4f  wadd[KADD];    // {fw1, fb1, 0.5*fw2, 0}              k=f*64+h
    __shared__ v4f  wpair[KPAIR];  // {pw1[p][0], pw1[p][1], pb1, 0.5*pw2} k=p*64+h
    __shared__ float msbuf[2 * NF];// mean[f], inv_std[f]
    __shared__ float caddbuf;      // sum(fb2) + sum(pb2)

    const int tid = threadIdx.x;
    for (int k = tid; k < KADD; k += 256) {
        int f = k >> 6, h = k & 63;
        v4f w; w.x = fw1[f * NH + h]; w.y = fb1[f * NH + h];
        w.z = 0.5f * fw2[f * NH + h]; w.w = 0.0f;
        wadd[k] = w;
    }
    for (int k = tid; k < KPAIR; k += 256) {
        int p = k >> 6, h = k & 63;
        v4f w; w.x = pw1[(p * 2 + 0) * NH + h]; w.y = pw1[(p * 2 + 1) * NH + h];
        w.z = pb1[p * NH + h]; w.w = 0.5f * pw2[p * NH + h];
        wpair[k] = w;
    }
    if (tid < NF) {
        msbuf[tid]      = raw_mean[tid];
        msbuf[NF + tid] = 1.0f / raw_std[tid];
    }
    if (tid == 0) {
        float c = 0.0f;
        for (int f = 0; f < NF; ++f) c += fb2[f];
        for (int p = 0; p < NP; ++p) c += pb2[p];
        caddbuf = c;
    }
    __syncthreads();

    // ---- per-wave 16-row tile ----
    const int lane = tid & 31;
    const int wave = tid >> 5;
    const int r    = lane & 15;   // row within tile (A-matrix M index)
    const int g    = lane >> 4;   // half-wave group: this lane's K slots = 2g, 2g+1
    const int tile = blockIdx.x * 8 + wave;      // 8 waves per block
    const int b0   = tile * 16;
    if (b0 >= nrows) return;                     // wave-uniform guard
    const int b = b0 + r;

    // load this row's 10 features (row byte offset b*40 is 8B aligned)
    const float2* xr2 = (const float2*)(x + (size_t)b * NF);
    float s[NF];
    #pragma unroll
    for (int f2 = 0; f2 < NF / 2; ++f2) {
        float2 v = xr2[f2];
        s[2 * f2 + 0] = v.x;
        s[2 * f2 + 1] = v.y;
    }
    // column 4: log(clip(x*10, 1e-6)), then normalize all features
    s[4] = logf(fmaxf(s[4] * 10.0f, 1e-6f));
    #pragma unroll
    for (int f = 0; f < NF; ++f) s[f] = (s[f] - msbuf[f]) * msbuf[NF + f];

    v8f c = {};              // 16x16 f32 accumulator; every column == row sum
    v2f ones; ones[0] = 1.0f; ones[1] = 1.0f;   // B matrix: all ones

    // ---- additive branch: k = f*64 + h, 160 WMMAs per tile ----
    #pragma unroll
    for (int f = 0; f < NF; ++f) {
        const float sf = s[f];
        const v4f* wf = &wadd[f * NH];
        #pragma unroll 1
        for (int hb = 0; hb < NH; hb += 4) {
            v4f w0 = wf[hb + 2 * g];       // k = hb+2g   -> A slot VGPR0
            v4f w1 = wf[hb + 2 * g + 1];   // k = hb+2g+1 -> A slot VGPR1
            v2f a;
            a[0] = wgelu(fmaf(sf, w0.x, w0.y), w0.z);
            a[1] = wgelu(fmaf(sf, w1.x, w1.y), w1.z);
            c = __builtin_amdgcn_wmma_f32_16x16x4_f32(
                    false, a, false, ones, (short)0, c, false, false);
        }
    }

    // ---- pair branch: 6 pairs, k = p*64 + h, 96 WMMAs per tile ----
    const int pi0[NP] = {0, 0, 0, 0, 1, 2};
    const int pi1[NP] = {4, 7, 9, 3, 4, 3};
    #pragma unroll
    for (int p = 0; p < NP; ++p) {
        const float s0 = s[pi0[p]];
        const float s1 = s[pi1[p]];
        const v4f* wp = &wpair[p * NH];
        #pragma unroll 1
        for (int hb = 0; hb < NH; hb += 4) {
            v4f w0 = wp[hb + 2 * g];
            v4f w1 = wp[hb + 2 * g + 1];
            v2f a;
            a[0] = wgelu(fmaf(s0, w0.x, fmaf(s1, w0.y, w0.z)), w0.w);
            a[1] = wgelu(fmaf(s0, w1.x, fmaf(s1, w1.y, w1.z)), w1.w);
            c = __builtin_amdgcn_wmma_f32_16x16x4_f32(
                    false, a, false, ones, (short)0, c, false, false);
        }
    }

    // ---- epilogue: C layout => lanes 0-15: c[j] = row j ; lanes 16-31: row j+8.
    // All columns identical, so lanes 0..7 write rows 0..7 and lanes 16..23
    // write rows 8..15.
    if ((lane & 15) < 8) {
        const int j = lane & 7;
        float v = c[0];
        #pragma unroll
        for (int q = 1; q < 8; ++q) if (j == q) v = c[q];
        out[b0 + j + 8 * g] = v + caddbuf;
    }
}

extern "C" void kernel_launch(void* const* d_in, const int* in_sizes, int n_in,
                              void* d_out, int out_size, void* d_ws, size_t ws_size,
                              hipStream_t stream) {
    (void)n_in; (void)d_ws; (void)ws_size; (void)out_size;
    const float* x        = (const float*)d_in[0];
    const float* raw_mean = (const float*)d_in[1];
    const float* raw_std  = (const float*)d_in[2];
    const float* fw1      = (const float*)d_in[3];
    const float* fb1      = (const float*)d_in[4];
    const float* fw2      = (const float*)d_in[5];
    const float* fb2      = (const float*)d_in[6];
    const float* pw1      = (const float*)d_in[7];
    const float* pb1      = (const float*)d_in[8];
    const float* pw2      = (const float*)d_in[9];
    const float* pb2      = (const float*)d_in[10];
    float* out = (float*)d_out;

    const int nrows = in_sizes[0] / NF;           // 524288
    const int tiles = (nrows + 15) / 16;          // 16 rows per wave
    const int blocks = (tiles + 7) / 8;           // 8 waves (256 thr) per block
    risk_model_kernel<<<blocks, 256, 0, stream>>>(
        x, raw_mean, raw_std, fw1, fb1, fw2, fb2, pw1, pb1, pw2, pb2,
        out, nrows);
}